// SynPool_46557445488795
// MI455X (gfx1250) — compile-verified
//
#include <hip/hip_runtime.h>
#include <math.h>

typedef float v2f __attribute__((ext_vector_type(2)));
typedef float v8f __attribute__((ext_vector_type(8)));

#define DIN   512
#define EDIM  512
#define NH    8
#define HEAD  64
#define SEQ   8192
#define BATCH 8
#define NROW  (BATCH * SEQ)
#define NSTEP 8
#define GAMMA   0.5f
#define SCALING 0.125f
#define EPSV    1e-5f

// ---------------- wave / block reductions (wave32) ----------------
__device__ __forceinline__ float wred_add(float v) {
#pragma unroll
    for (int off = 16; off; off >>= 1) v += __shfl_xor(v, off, 32);
    return v;
}
__device__ __forceinline__ float wred_max(float v) {
#pragma unroll
    for (int off = 16; off; off >>= 1) v = fmaxf(v, __shfl_xor(v, off, 32));
    return v;
}
// blockDim.x == 256 (8 waves)
__device__ __forceinline__ float bred_add(float v, float* red) {
    int lane = threadIdx.x & 31, wid = threadIdx.x >> 5;
    v = wred_add(v);
    if (lane == 0) red[wid] = v;
    __syncthreads();
    if (wid == 0) {
        float t = (lane < 8) ? red[lane] : 0.0f;
        t = wred_add(t);
        if (lane == 0) red[0] = t;
    }
    __syncthreads();
    float r = red[0];
    __syncthreads();
    return r;
}
__device__ __forceinline__ float bred_max(float v, float* red) {
    int lane = threadIdx.x & 31, wid = threadIdx.x >> 5;
    v = wred_max(v);
    if (lane == 0) red[wid] = v;
    __syncthreads();
    if (wid == 0) {
        float t = (lane < 8) ? red[lane] : -3.4e38f;
        t = wred_max(t);
        if (lane == 0) red[0] = t;
    }
    __syncthreads();
    float r = red[0];
    __syncthreads();
    return r;
}

// ---------------- kernel 1: q projection + fold into W_k ----------------
// qs[e] = SCALING * (q_param . W_q[e,:] + b_q[e])
// wq_eff[h][c] = sum_d qs[h*64+d] * W_kv[h*64+d][c]   (k-half of W_kv)
// qb[h]       = sum_d qs[h*64+d] * b_kv[h*64+d]
__global__ void k_qprep(const float* __restrict__ W_kv, const float* __restrict__ b_kv,
                        const float* __restrict__ W_q,  const float* __restrict__ b_q,
                        const float* __restrict__ q_param,
                        float* __restrict__ wq_eff, float* __restrict__ qb) {
    __shared__ float qp[DIN];
    __shared__ float qs[EDIM];
    int t = threadIdx.x; // 512
    qp[t] = q_param[t];
    __syncthreads();
    float acc = 0.0f;
    const float* wr = W_q + (size_t)t * DIN;
    for (int c = 0; c < DIN; ++c) acc += qp[c] * wr[c];
    qs[t] = (acc + b_q[t]) * SCALING;
    __syncthreads();
    // wq_eff: column c == t, all 8 heads
    for (int h = 0; h < NH; ++h) {
        float a = 0.0f;
        for (int d = 0; d < HEAD; ++d)
            a += qs[h * HEAD + d] * W_kv[(size_t)(h * HEAD + d) * DIN + t];
        wq_eff[h * DIN + t] = a;
    }
    if (t < NH) {
        float a = 0.0f;
        for (int d = 0; d < HEAD; ++d) a += qs[t * HEAD + d] * b_kv[t * HEAD + d];
        qb[t] = a;
    }
}

// ---------------- kernel 2: zero scratch ----------------
__global__ void k_zero(float* __restrict__ p, int n) {
    int i = blockIdx.x * blockDim.x + threadIdx.x;
    if (i < n) p[i] = 0.0f;
}

// ---------------- kernel 3: fused LayerNorm + 8-head scores ----------------
// one wave per row; grid = 8192 blocks x 256 threads
__global__ void k_ln_scores(const float* __restrict__ x,
                            const float* __restrict__ ln_w, const float* __restrict__ ln_b,
                            const float* __restrict__ wq_eff, const float* __restrict__ qb,
                            float* __restrict__ mu_out, float* __restrict__ rstd_out,
                            float* __restrict__ scores) {
    __shared__ float s_wq[NH * DIN];
    __shared__ float s_lnw[DIN];
    __shared__ float s_lnb[DIN];
    for (int i = threadIdx.x; i < NH * DIN; i += blockDim.x) s_wq[i] = wq_eff[i];
    for (int i = threadIdx.x; i < DIN; i += blockDim.x) { s_lnw[i] = ln_w[i]; s_lnb[i] = ln_b[i]; }
    __syncthreads();

    int wid = threadIdx.x >> 5, lane = threadIdx.x & 31;
    int row = blockIdx.x * 8 + wid;            // 0..65535
    const float* xr = x + (size_t)row * DIN;

    float vals[16];
    float s = 0.0f, ss = 0.0f;
#pragma unroll
    for (int j = 0; j < 4; ++j) {
        float4 t = ((const float4*)xr)[lane + 32 * j];
        vals[4 * j + 0] = t.x; vals[4 * j + 1] = t.y;
        vals[4 * j + 2] = t.z; vals[4 * j + 3] = t.w;
        s  += t.x + t.y + t.z + t.w;
        ss += t.x * t.x + t.y * t.y + t.z * t.z + t.w * t.w;
    }
    s  = wred_add(s);
    ss = wred_add(ss);
    float mu   = s * (1.0f / DIN);
    float var  = ss * (1.0f / DIN) - mu * mu;
    float rstd = rsqrtf(var + EPSV);

    float acc[NH];
#pragma unroll
    for (int h = 0; h < NH; ++h) acc[h] = 0.0f;
#pragma unroll
    for (int j = 0; j < 4; ++j) {
#pragma unroll
        for (int i = 0; i < 4; ++i) {
            int c = (lane + 32 * j) * 4 + i;
            float d = (vals[4 * j + i] - mu) * rstd * s_lnw[c] + s_lnb[c];
#pragma unroll
            for (int h = 0; h < NH; ++h) acc[h] += d * s_wq[h * DIN + c];
        }
    }
#pragma unroll
    for (int h = 0; h < NH; ++h) acc[h] = wred_add(acc[h]);

    if (lane == 0) {
        mu_out[row]   = mu;
        rstd_out[row] = rstd;
        int b  = row >> 13;         // /8192
        int sl = row & (SEQ - 1);
#pragma unroll
        for (int h = 0; h < NH; ++h)
            scores[(size_t)((b * NH + h)) * SEQ + sl] = acc[h] + qb[h];
    }
}

// ---------------- kernel 4: softmax + 8x syn + row-sum ----------------
// grid = 64 blocks (b*8+h) x 256 threads; 32 values per thread in registers
__global__ void k_softmax_syn(const float* __restrict__ scores,
                              float* __restrict__ attn_out,   // d_out + 4096
                              float* __restrict__ attn16,     // zero-padded [b][16][S]
                              float* __restrict__ attnsum) {
    __shared__ float red[8];
    int r = blockIdx.x;              // b*8 + h
    int t = threadIdx.x;
    const float* src = scores + (size_t)r * SEQ;

    float v[32];
#pragma unroll
    for (int j = 0; j < 32; ++j) v[j] = src[t + 256 * j];

    float m = -3.4e38f;
#pragma unroll
    for (int j = 0; j < 32; ++j) m = fmaxf(m, v[j]);
    m = bred_max(m, red);

    float sum = 0.0f;
#pragma unroll
    for (int j = 0; j < 32; ++j) { v[j] = expf(v[j] - m); sum += v[j]; }
    sum = bred_add(sum, red);
    float inv = 1.0f / sum;
#pragma unroll
    for (int j = 0; j < 32; ++j) v[j] *= inv;

    for (int it = 0; it < NSTEP; ++it) {
        float ssq = 0.0f;
#pragma unroll
        for (int j = 0; j < 32; ++j) ssq += v[j] * v[j];
        ssq = bred_add(ssq, red);
        float invn = rsqrtf(ssq);
#pragma unroll
        for (int j = 0; j < 32; ++j) {
            float u = v[j] * invn;
            v[j] = u + GAMMA * u * (u * u - 1.0f);
        }
    }

    float asum = 0.0f;
#pragma unroll
    for (int j = 0; j < 32; ++j) asum += v[j];
    asum = bred_add(asum, red);

    int b = r >> 3, h = r & 7;
    float* o1 = attn_out + (size_t)r * SEQ;
    float* o2 = attn16 + ((size_t)(b * 16 + h)) * SEQ;
#pragma unroll
    for (int j = 0; j < 32; ++j) { o1[t + 256 * j] = v[j]; o2[t + 256 * j] = v[j]; }
    if (t == 0) attnsum[r] = asum;
}

// ---------------- kernel 5: pooled[b][h][c] via f32 WMMA ----------------
// D[16x16] += A[16x4](attn, M=head) x B[4x16](LN(x), N=channel), K = seq in steps of 4.
// grid = (32 c-tiles, 16 s-chunks, 8 b), block = 32 threads (one wave).
__global__ void k_pool_wmma(const float* __restrict__ x,
                            const float* __restrict__ ln_w, const float* __restrict__ ln_b,
                            const float* __restrict__ mu,  const float* __restrict__ rstd,
                            const float* __restrict__ attn16,
                            float* __restrict__ pooled /* [8][16][512] */) {
    int lane = threadIdx.x;
    int half = lane >> 4;            // 0: K=0/1, 1: K=2/3
    int nl   = lane & 15;
    int b    = blockIdx.z;
    int c    = blockIdx.x * 16 + nl;
    int s0   = blockIdx.y * (SEQ / 16);

    const float* xb = x + (size_t)b * SEQ * DIN;
    const float* at = attn16 + (size_t)b * 16 * SEQ;
    float lnw_c = ln_w[c], lnb_c = ln_b[c];
    int m  = lane & 15;              // A-matrix row (head, 8..15 are zero-padded)
    int k0 = half * 2;

    v8f acc = {};
    for (int s = s0; s < s0 + SEQ / 16; s += 4) {
        int r0 = s + k0;
        int r1 = r0 + 1;
        int g0 = b * SEQ + r0;
        int g1 = g0 + 1;

        v2f a;
        a.x = at[(size_t)m * SEQ + r0];
        a.y = at[(size_t)m * SEQ + r1];

        float x0 = xb[(size_t)r0 * DIN + c];
        float x1 = xb[(size_t)r1 * DIN + c];
        v2f bf;
        bf.x = (x0 - mu[g0]) * rstd[g0] * lnw_c + lnb_c;
        bf.y = (x1 - mu[g1]) * rstd[g1] * lnw_c + lnb_c;

        // prefetch a few rows ahead (clamped, branch-free)
        int pr = r0 + 8; pr = (pr < SEQ) ? pr : r0;
        __builtin_prefetch(&xb[(size_t)pr * DIN + c], 0, 1);

        acc = __builtin_amdgcn_wmma_f32_16x16x4_f32(
            false, a, false, bf, (short)0, acc, false, false);
    }

    float* pb = pooled + (size_t)b * 16 * DIN;
#pragma unroll
    for (int i = 0; i < 8; ++i) {
        int M = i + half * 8;        // D layout: vgpr i -> rows i / i+8
        atomicAdd(&pb[(size_t)M * DIN + blockIdx.x * 16 + nl], acc[i]);
    }
}

// ---------------- kernel 6: ctx[b][e] = pooled[b][h] . W_v[e] + attnsum*b_kv ----------------
__global__ void k_ctx(const float* __restrict__ pooled, const float* __restrict__ W_kv,
                      const float* __restrict__ b_kv, const float* __restrict__ attnsum,
                      float* __restrict__ ctx) {
    int b = blockIdx.x, e = threadIdx.x; // 512
    int h = e >> 6;
    const float* pr = pooled + ((size_t)b * 16 + h) * DIN;
    const float* wr = W_kv + (size_t)(EDIM + e) * DIN;   // v-half of W_kv
    float a = 0.0f;
    for (int cc = 0; cc < DIN; ++cc) a += pr[cc] * wr[cc];
    ctx[b * EDIM + e] = a + attnsum[b * NH + h] * b_kv[EDIM + e];
}

// ---------------- kernel 7: out[b][o] = ctx[b] . W_out[o] + b_out[o] ----------------
__global__ void k_out(const float* __restrict__ ctx, const float* __restrict__ W_out,
                      const float* __restrict__ b_out, float* __restrict__ out) {
    __shared__ float cs[EDIM];
    int b = blockIdx.x, o = threadIdx.x; // 512
    cs[o] = ctx[b * EDIM + o];
    __syncthreads();
    const float* wr = W_out + (size_t)o * EDIM;
    float a = 0.0f;
    for (int e = 0; e < EDIM; ++e) a += cs[e] * wr[e];
    out[b * EDIM + o] = a + b_out[o];
}

// ---------------- launcher ----------------
extern "C" void kernel_launch(void* const* d_in, const int* in_sizes, int n_in,
                              void* d_out, int out_size, void* d_ws, size_t ws_size,
                              hipStream_t stream) {
    const float* x       = (const float*)d_in[0];
    const float* ln_w    = (const float*)d_in[1];
    const float* ln_b    = (const float*)d_in[2];
    const float* W_kv    = (const float*)d_in[3];
    const float* b_kv    = (const float*)d_in[4];
    const float* W_q     = (const float*)d_in[5];
    const float* b_q     = (const float*)d_in[6];
    const float* q_param = (const float*)d_in[7];
    const float* W_out   = (const float*)d_in[8];
    const float* b_out   = (const float*)d_in[9];

    float* out      = (float*)d_out;            // 4096 floats
    float* attn_out = out + BATCH * EDIM;       // 524288 floats

    // workspace layout (floats)
    float* ws      = (float*)d_ws;
    float* wq_eff  = ws;                                  // 4096
    float* qb      = wq_eff + NH * DIN;                   // 8
    float* attnsum = qb + 16;                             // 64
    float* ctx     = attnsum + 64;                        // 4096
    float* mu      = ctx + BATCH * EDIM;                  // 65536
    float* rstd    = mu + NROW;                           // 65536
    float* scores  = rstd + NROW;                         // 524288
    float* pooled  = scores + BATCH * NH * SEQ;           // 65536  (zeroed)
    float* attn16  = pooled + BATCH * 16 * DIN;           // 1048576 (zeroed)

    k_qprep<<<1, 512, 0, stream>>>(W_kv, b_kv, W_q, b_q, q_param, wq_eff, qb);

    int nz = BATCH * 16 * DIN + BATCH * 16 * SEQ;         // pooled + attn16 (contiguous)
    k_zero<<<(nz + 255) / 256, 256, 0, stream>>>(pooled, nz);

    k_ln_scores<<<NROW / 8, 256, 0, stream>>>(x, ln_w, ln_b, wq_eff, qb, mu, rstd, scores);

    k_softmax_syn<<<BATCH * NH, 256, 0, stream>>>(scores, attn_out, attn16, attnsum);

    k_pool_wmma<<<dim3(DIN / 16, 16, BATCH), 32, 0, stream>>>(
        x, ln_w, ln_b, mu, rstd, attn16, pooled);

    k_ctx<<<BATCH, EDIM, 0, stream>>>(pooled, W_kv, b_kv, attnsum, ctx);
    k_out<<<BATCH, EDIM, 0, stream>>>(ctx, W_out, b_out, out);
}